// CfCDecoder_67989332295970
// MI455X (gfx1250) — compile-verified
//
#include <hip/hip_runtime.h>
#include <hip/hip_bf16.h>

#define B_   64
#define L_   128
#define TE_  256
#define CIN_ 256
#define H_   512
#define OUT_ 128

typedef __attribute__((ext_vector_type(2))) float v2f;
typedef __attribute__((ext_vector_type(8))) float v8f;

__device__ __forceinline__ float sigm(float x) { return 1.0f / (1.0f + expf(-x)); }

__global__ void zero_kernel(float* __restrict__ p, int n) {
  int i = blockIdx.x * blockDim.x + threadIdx.x;
  if (i < n) p[i] = 0.0f;
}

// Y[M,N] = act( concat(X1,X2)_rowmajor @ concat(W1,W2)^T + bias )
//   X row m = X1[m*s1 .. +K1) ++ X2[m*s2 .. +K2)
//   W row n = W1[n*KW1 .. +KW1) ++ W2[n*KW2 .. +KW2)   (K1+K2 == KW1+KW2, all even)
// grid = (N/128, M/16), block = 128 (4 waves; wave w owns a 16x32 tile: cols
// (blockIdx.x*4+w)*32 .. +32). Two accumulators per wave share one A fragment.
// fp32 WMMA: A 16x4 = 2 VGPRs/lane (lanes 0-15: K=k0,k0+1 ; lanes 16-31: K=k0+2,k0+3)
__global__ void gemm_wmma(const float* __restrict__ X1, int s1, int K1,
                          const float* __restrict__ X2, int s2, int K2,
                          const float* __restrict__ W1, int KW1,
                          const float* __restrict__ W2, int KW2,
                          const float* __restrict__ bias,
                          float* __restrict__ Y, int ldy, int act)
{
  const int wave = threadIdx.x >> 5;
  const int lane = threadIdx.x & 31;
  const int half = lane >> 4;   // which K-pair this lane carries
  const int l16  = lane & 15;
  const int N0 = (blockIdx.x * 4 + wave) * 32;  // first of two 16-col tiles
  const int N1 = N0 + 16;
  const int M0 = blockIdx.y * 16;
  const int K  = K1 + K2;
  const int m  = M0 + l16;      // A row for this lane
  const int n0 = N0 + l16;      // B column (tile 0) for this lane
  const int n1 = N1 + l16;      // B column (tile 1) for this lane

  v8f acc0 = {};
  v8f acc1 = {};

  // Split K into segments so X-side and W-side base pointers are fixed per
  // segment (no per-iteration selects); pointers then advance by 4 floats.
  const int sA = (K1 < KW1) ? K1 : KW1;
  const int sB = (K1 < KW1) ? KW1 : K1;
  int bnd[4] = { 0, sA, sB, K };

  #pragma unroll
  for (int si = 0; si < 3; ++si) {
    const int lo = bnd[si], hi = bnd[si + 1];
    if (hi <= lo) continue;
    const float* pa  = (lo < K1)  ? (X1 + (size_t)m  * s1  + lo)
                                  : (X2 + (size_t)m  * s2  + (lo - K1));
    const float* pb0 = (lo < KW1) ? (W1 + (size_t)n0 * KW1 + lo)
                                  : (W2 + (size_t)n0 * KW2 + (lo - KW1));
    const float* pb1 = (lo < KW1) ? (W1 + (size_t)n1 * KW1 + lo)
                                  : (W2 + (size_t)n1 * KW2 + (lo - KW1));
    pa  += half * 2;
    pb0 += half * 2;
    pb1 += half * 2;
    #pragma unroll 8
    for (int k = lo; k < hi; k += 4) {
      v2f a  = *(const v2f*)pa;     // 8B-aligned: strides/splits all even
      v2f b0 = *(const v2f*)pb0;
      v2f b1 = *(const v2f*)pb1;
      acc0 = __builtin_amdgcn_wmma_f32_16x16x4_f32(false, a, false, b0,
                                                   (short)0, acc0, false, false);
      acc1 = __builtin_amdgcn_wmma_f32_16x16x4_f32(false, a, false, b1,
                                                   (short)0, acc1, false, false);
      pa += 4; pb0 += 4; pb1 += 4;
    }
  }

  const float bv0 = bias ? bias[n0] : 0.0f;
  const float bv1 = bias ? bias[n1] : 0.0f;
  const int rowbase = M0 + half * 8;
  #pragma unroll
  for (int r = 0; r < 8; ++r) {
    float v0 = acc0[r] + bv0;
    float v1 = acc1[r] + bv1;
    if (act) {
      v0 = (v0 > 0.0f) ? v0 : 0.01f * v0;   // leaky_relu
      v1 = (v1 > 0.0f) ? v1 : 0.01f * v1;
    }
    Y[(size_t)(rowbase + r) * ldy + n0] = v0;
    Y[(size_t)(rowbase + r) * ldy + n1] = v1;
  }
}

// One block per batch row; 256 threads (1 thread per encoder timestep).
__global__ void attn_kernel(const float* __restrict__ proj_dec,  // B x H
                            const float* __restrict__ proj_enc,  // B x TE x H
                            const float* __restrict__ w_score,   // H
                            const float* __restrict__ h_en,      // B x TE x H
                            float* __restrict__ ctx)             // B x H
{
  __shared__ float pd[H_];
  __shared__ float buf[TE_];
  __shared__ float sc[TE_];
  const int b = blockIdx.x;
  const int tid = threadIdx.x;
  for (int h = tid; h < H_; h += TE_) pd[h] = proj_dec[b * H_ + h];
  __syncthreads();

  const float* pe = proj_enc + ((size_t)b * TE_ + tid) * H_;
  float e = 0.0f;
  for (int h = 0; h < H_; ++h) e += tanhf(pd[h] + pe[h]) * w_score[h];

  // softmax over TE_ values, one per thread
  buf[tid] = e; __syncthreads();
  for (int s = TE_ / 2; s > 0; s >>= 1) {
    if (tid < s) buf[tid] = fmaxf(buf[tid], buf[tid + s]);
    __syncthreads();
  }
  const float mx = buf[0]; __syncthreads();
  const float ex = expf(e - mx);
  buf[tid] = ex; __syncthreads();
  for (int s = TE_ / 2; s > 0; s >>= 1) {
    if (tid < s) buf[tid] += buf[tid + s];
    __syncthreads();
  }
  const float inv = 1.0f / buf[0];
  sc[tid] = ex * inv;
  __syncthreads();

  // context: ctx[b,h] = sum_t score[t] * h_en[b,t,h]
  for (int h = tid; h < H_; h += TE_) {
    float acc = 0.0f;
    const float* he = h_en + (size_t)b * TE_ * H_ + h;
    for (int t = 0; t < TE_; ++t) acc += sc[t] * he[(size_t)t * H_];
    ctx[b * H_ + h] = acc;
  }
}

// ncps LSTMCell gates (forget bias +1); z layout [i | ig | fg | og], updates c in place.
__global__ void lstm_kernel(const float* __restrict__ z,
                            float* __restrict__ c,
                            float* __restrict__ h_lstm)
{
  int i = blockIdx.x * blockDim.x + threadIdx.x;
  if (i >= B_ * H_) return;
  const int b = i >> 9, n = i & (H_ - 1);
  const float* zr = z + (size_t)b * 4 * H_;
  const float i_ = zr[n];
  const float ig = zr[H_ + n];
  const float fg = zr[2 * H_ + n];
  const float og = zr[3 * H_ + n];
  const float nc = c[i] * sigm(fg + 1.0f) + tanhf(i_) * sigm(ig);
  c[i] = nc;
  h_lstm[i] = tanhf(nc) * sigm(og);
}

// CfC combine; g4 layout [ff1 | ff2 | ta | tb], writes new hidden state h.
__global__ void cfc_kernel(const float* __restrict__ g4, float* __restrict__ h)
{
  int i = blockIdx.x * blockDim.x + threadIdx.x;
  if (i >= B_ * H_) return;
  const int b = i >> 9, n = i & (H_ - 1);
  const float* g = g4 + (size_t)b * 4 * H_;
  const float f1 = tanhf(g[n]);
  const float f2 = tanhf(g[H_ + n]);
  const float ti = sigm(g[2 * H_ + n] * 1.0f + g[3 * H_ + n]);  // ts = 1
  h[i] = f1 * (1.0f - ti) + ti * f2;
}

extern "C" void kernel_launch(void* const* d_in, const int* in_sizes, int n_in,
                              void* d_out, int out_size, void* d_ws, size_t ws_size,
                              hipStream_t stream) {
  const float* x       = (const float*)d_in[0];   // B x L x CIN
  const float* h_en    = (const float*)d_in[1];   // B x TE x H
  const float* W_dec   = (const float*)d_in[2];   // H x 2H
  const float* W_enc   = (const float*)d_in[3];   // H x H
  const float* b_enc   = (const float*)d_in[4];
  const float* w_score = (const float*)d_in[5];   // 1 x H
  const float* W_yattn = (const float*)d_in[6];   // CIN x (H+CIN)
  const float* b_yattn = (const float*)d_in[7];
  const float* W_ih    = (const float*)d_in[8];   // 4H x CIN
  const float* b_ih    = (const float*)d_in[9];
  const float* W_hh    = (const float*)d_in[10];  // 4H x H
  const float* W_ff1   = (const float*)d_in[11];  // H x (CIN+H)
  const float* b_ff1   = (const float*)d_in[12];
  const float* W_ff2   = (const float*)d_in[13];
  const float* b_ff2   = (const float*)d_in[14];
  const float* W_ta    = (const float*)d_in[15];
  const float* b_ta    = (const float*)d_in[16];
  const float* W_tb    = (const float*)d_in[17];
  const float* b_tb    = (const float*)d_in[18];
  const float* W_o1    = (const float*)d_in[19];  // H x 2H
  const float* b_o1    = (const float*)d_in[20];
  const float* W_o2    = (const float*)d_in[21];  // OUT x H
  const float* b_o2    = (const float*)d_in[22];

  float* ws = (float*)d_ws;
  float* proj_enc = ws; ws += (size_t)B_ * TE_ * H_;  // 32 MB - L2 resident
  float* h   = ws; ws += B_ * H_;
  float* c   = ws; ws += B_ * H_;
  float* ctx = ws; ws += B_ * H_;
  float* xh  = ws; ws += B_ * CIN_;
  float* z   = ws; ws += B_ * 4 * H_;
  float* g4  = ws; ws += B_ * 4 * H_;
  float* hl  = ws; ws += B_ * H_;
  float* hid = ws; ws += B_ * H_;
  float* pd  = ws; ws += B_ * H_;

  auto gemm = [&](const float* X1, int s1, int K1,
                  const float* X2, int s2, int K2,
                  const float* Wa, int KW1, const float* Wb, int KW2,
                  const float* bias, float* Y, int ldy, int M, int N, int act) {
    dim3 grid(N / 128, M / 16);
    gemm_wmma<<<grid, dim3(128), 0, stream>>>(X1, s1, K1, X2, s2, K2,
                                              Wa, KW1, Wb, KW2, bias, Y, ldy, act);
  };

  // h0 = c0 = 0 (h and c are adjacent in workspace)
  zero_kernel<<<(2 * B_ * H_ + 255) / 256, 256, 0, stream>>>(h, 2 * B_ * H_);

  // proj_enc = h_en @ W_enc^T + b_enc  (M = B*TE = 16384, K = N = 512)
  gemm(h_en, H_, H_, nullptr, 0, 0, W_enc, H_, nullptr, 0,
       b_enc, proj_enc, H_, B_ * TE_, H_, 0);

  for (int t = 0; t < L_; ++t) {
    // proj_dec = [h, c] @ W_dec^T
    gemm(h, H_, H_, c, H_, H_, W_dec, 2 * H_, nullptr, 0,
         nullptr, pd, H_, B_, H_, 0);
    // attention scores + softmax + context
    attn_kernel<<<B_, TE_, 0, stream>>>(pd, proj_enc, w_score, h_en, ctx);
    // x_hat = [x_t, ctx] @ W_yattn^T + b_yattn
    gemm(x + (size_t)t * CIN_, L_ * CIN_, CIN_, ctx, H_, H_,
         W_yattn, CIN_ + H_, nullptr, 0, b_yattn, xh, CIN_, B_, CIN_, 0);
    // z = x_hat @ W_ih^T + b_ih + h @ W_hh^T   (concat on both X and W sides)
    gemm(xh, CIN_, CIN_, h, H_, H_, W_ih, CIN_, W_hh, H_,
         b_ih, z, 4 * H_, B_, 4 * H_, 0);
    lstm_kernel<<<(B_ * H_ + 255) / 256, 256, 0, stream>>>(z, c, hl);
    // four CfC projections from xx = [x_hat, h_lstm] into quadrants of g4
    const float* Wff[4] = { W_ff1, W_ff2, W_ta, W_tb };
    const float* bff[4] = { b_ff1, b_ff2, b_ta, b_tb };
    for (int q = 0; q < 4; ++q)
      gemm(xh, CIN_, CIN_, hl, H_, H_, Wff[q], CIN_ + H_, nullptr, 0,
           bff[q], g4 + q * H_, 4 * H_, B_, H_, 0);
    cfc_kernel<<<(B_ * H_ + 255) / 256, 256, 0, stream>>>(g4, h);
  }

  // hid = leaky_relu([h_out, c_last] @ W_o1^T + b_o1); c_last = last attention context
  gemm(h, H_, H_, ctx, H_, H_, W_o1, 2 * H_, nullptr, 0,
       b_o1, hid, H_, B_, H_, 1);
  // out = hid @ W_o2^T + b_o2
  float* out = (float*)d_out;
  gemm(hid, H_, H_, nullptr, 0, 0, W_o2, H_, nullptr, 0,
       b_o2, out, OUT_, B_, OUT_, 0);

  // tuple outputs: (out, h_state, c_state) flat-concatenated
  hipMemcpyAsync(out + B_ * OUT_, h, (size_t)B_ * H_ * sizeof(float),
                 hipMemcpyDeviceToDevice, stream);
  hipMemcpyAsync(out + B_ * OUT_ + B_ * H_, c, (size_t)B_ * H_ * sizeof(float),
                 hipMemcpyDeviceToDevice, stream);
}